// Model_50379966382548
// MI455X (gfx1250) — compile-verified
//
#include <hip/hip_runtime.h>
#include <hip/hip_bf16.h>

#define WAVE_SZ 32
#define WPB 8                          // waves per block (wave32!)
#define BLOCK_SZ (WAVE_SZ * WPB)       // 256 threads = 8 waves on CDNA5
#define MAXSEG 512                     // mid-path max segment length
#define CHUNKS (MAXSEG / WAVE_SZ)      // 16

#if __has_builtin(__builtin_amdgcn_global_load_async_to_lds_b32)
#define HAS_ASYNC_LDS 1
#else
#define HAS_ASYNC_LDS 0
#endif

typedef __attribute__((address_space(1))) int as1_int;   // global (prints as __device__ int*)
typedef __attribute__((address_space(3))) int as3_int;   // LDS    (prints as __shared__ int*)

__device__ __forceinline__ float wave_max32(float v) {
#pragma unroll
  for (int off = 16; off > 0; off >>= 1)
    v = fmaxf(v, __shfl_xor(v, off, WAVE_SZ));
  return v;
}

__device__ __forceinline__ float wave_sum32(float v) {
#pragma unroll
  for (int off = 16; off > 0; off >>= 1)
    v += __shfl_xor(v, off, WAVE_SZ);
  return v;
}

extern "C" __global__ __launch_bounds__(BLOCK_SZ)
void seg_softmax_kernel(const long long* __restrict__ row_ptr,
                        const float* __restrict__ scores,
                        float* __restrict__ out,
                        int num_nodes)
{
#if HAS_ASYNC_LDS
  __shared__ float stage[WPB * MAXSEG];   // 16 KB of the 320 KB/WGP pool
#endif
  const int lane = threadIdx.x & (WAVE_SZ - 1);
  const int wave = threadIdx.x >> 5;
  // wave-uniform node id; readfirstlane lets the compiler scalarize row_ptr loads
  const int node = __builtin_amdgcn_readfirstlane((int)blockIdx.x * WPB + wave);
  if (node >= num_nodes) return;

  const long long s0 = row_ptr[node];
  const long long s1 = row_ptr[node + 1];
  const int len = (int)(s1 - s0);
  if (len <= 0) return;                  // empty CSR row

  const float* __restrict__ seg  = scores + s0;
  float*       __restrict__ oseg = out + s0;
  const float NEG_INF = -__builtin_inff();

  if (len <= WAVE_SZ) {
    // ---- Hot path (~63% of nodes: Exp(32) gap distribution): one chunk,
    // pure register flow, no LDS, no loops.
    const float x = (lane < len) ? seg[lane] : NEG_INF;
    const float m = wave_max32(x);
    const float e = __expf(x - m);           // exp(-inf)=0 for masked lanes
    const float s = wave_sum32(e);
    if (lane < len)
      __builtin_nontemporal_store(e * (1.0f / s), &oseg[lane]);
  } else if (len <= MAXSEG) {
    const int nch = (len + WAVE_SZ - 1) >> 5;
    float x[CHUNKS];

#if HAS_ASYNC_LDS
    // CDNA5 async copy: global -> LDS, tracked by ASYNCcnt. Branchless issue:
    // clamp the tail indices to len-1; colliding LDS writes carry identical data.
    float* wstage = &stage[wave * MAXSEG];
    for (int c = 0; c < nch; ++c) {
      int idx = (c << 5) + lane;
      idx = (idx < len) ? idx : (len - 1);
      __builtin_amdgcn_global_load_async_to_lds_b32(
          (as1_int*)(void*)(const void*)(seg + idx),
          (as3_int*)(void*)(wstage + idx),
          0, 0);
    }
#if __has_builtin(__builtin_amdgcn_s_wait_asynccnt)
    __builtin_amdgcn_s_wait_asynccnt(0);
#else
    asm volatile("s_wait_asynccnt 0" ::: "memory");
#endif
#pragma unroll
    for (int c = 0; c < CHUNKS; ++c) {
      if (c < nch) {
        const int idx = (c << 5) + lane;
        x[c] = (idx < len) ? wstage[idx] : NEG_INF;
      }
    }
#else
    // Fallback: straight coalesced register staging (single global read).
#pragma unroll
    for (int c = 0; c < CHUNKS; ++c) {
      if (c < nch) {
        const int idx = (c << 5) + lane;
        x[c] = (idx < len) ? seg[idx] : NEG_INF;
      }
    }
#endif

    float m = NEG_INF;
#pragma unroll
    for (int c = 0; c < CHUNKS; ++c) if (c < nch) m = fmaxf(m, x[c]);
    m = wave_max32(m);

    float s = 0.f;
#pragma unroll
    for (int c = 0; c < CHUNKS; ++c) {
      if (c < nch) {
        const float e = __expf(x[c] - m);
        x[c] = e;
        s += e;
      }
    }
    s = wave_sum32(s);
    const float inv = 1.0f / s;

#pragma unroll
    for (int c = 0; c < CHUNKS; ++c) {
      if (c < nch) {
        const int idx = (c << 5) + lane;
        if (idx < len)
          __builtin_nontemporal_store(x[c] * inv, &oseg[idx]);
      }
    }
  } else {
    // Rare long-segment path (expected ~0 segments > 512): multi-pass,
    // score re-reads hit the 192 MB L2 (whole edge array is 128 MB).
    float m = NEG_INF;
    for (int idx = lane; idx < len; idx += WAVE_SZ) m = fmaxf(m, seg[idx]);
    m = wave_max32(m);
    float s = 0.f;
    for (int idx = lane; idx < len; idx += WAVE_SZ) s += __expf(seg[idx] - m);
    s = wave_sum32(s);
    const float inv = 1.0f / s;
    for (int idx = lane; idx < len; idx += WAVE_SZ)
      __builtin_nontemporal_store(__expf(seg[idx] - m) * inv, &oseg[idx]);
  }
}

extern "C" void kernel_launch(void* const* d_in, const int* in_sizes, int n_in,
                              void* d_out, int out_size, void* d_ws, size_t ws_size,
                              hipStream_t stream) {
  (void)n_in; (void)out_size; (void)d_ws; (void)ws_size;
  const long long* row_ptr = (const long long*)d_in[0];  // int64 per reference
  const float*     scores  = (const float*)d_in[1];      // float32
  float*           out     = (float*)d_out;              // float32

  const int num_nodes = in_sizes[0] - 1;                 // row_ptr has N+1 entries
  const int blocks = (num_nodes + WPB - 1) / WPB;        // one wave per node
  seg_softmax_kernel<<<blocks, BLOCK_SZ, 0, stream>>>(row_ptr, scores, out, num_nodes);
}